// MHA_88330297409647
// MI455X (gfx1250) — compile-verified
//
#include <hip/hip_runtime.h>
#include <math.h>

// ---------------- problem constants (match reference) ----------------
#define D_MODEL 2048
#define NHEADS  16
#define DK      128
#define SEQ     2048
#define BATCH   2
#define MROWS   (BATCH * SEQ)          // 4096 flattened (b,s) rows
#define LOG_THETA 9.210340371976184f   // ln(10000)

typedef __attribute__((ext_vector_type(2))) float v2f;
typedef __attribute__((ext_vector_type(8))) float v8f;

// Exact-fp32 matrix op: D(16x16) = A(16x4) * B(4x16) + C
// A layout: lane L holds row M=L&15; VGPR0 = K(0|2), VGPR1 = K(1|3) by lane-half.
// B layout: lane L holds col N=L&15; VGPR0 = K(0|2), VGPR1 = K(1|3) by lane-half.
// C layout: VGPR i, lanes 0-15 -> (M=i, N=lane); lanes 16-31 -> (M=i+8, N=lane-16).
__device__ __forceinline__ v8f wmma_f32(v2f a, v2f b, v8f c) {
  return __builtin_amdgcn_wmma_f32_16x16x4_f32(false, a, false, b,
                                               (short)0, c, false, false);
}

__device__ __forceinline__ v8f vzero8() {
  v8f z = {0.f, 0.f, 0.f, 0.f, 0.f, 0.f, 0.f, 0.f};
  return z;
}

// =====================================================================
// Kernel 1: fused QKV projection + RoPE epilogue.
//   out[m, n] = sum_d x[m, d] * W[n, d]          (NT layout: both contiguous in d)
//   grid = (MROWS/16, (D_MODEL/16)/4, 3), block = 32 (one wave)
//   Each wave computes a 16x64 slab (4 N-tiles, A fragment reused 4x).
//   Q/K get RoPE applied in-register (adjacent lanes hold the even/odd pair).
// =====================================================================
__global__ __launch_bounds__(32)
void mha_qkv_rope(const float* __restrict__ x, const int* __restrict__ tpos,
                  const float* __restrict__ WQ, const float* __restrict__ WK,
                  const float* __restrict__ WV,
                  float* __restrict__ Qo, float* __restrict__ Ko,
                  float* __restrict__ Vo) {
  const int lane = threadIdx.x;
  const int hi = lane >> 4;      // K-half selector
  const int lo = lane & 15;      // row (A) / col (B,C) within tile
  const int m0 = blockIdx.x * 16;
  const int n0 = blockIdx.y * 64;
  const int mat = blockIdx.z;    // 0=Q 1=K 2=V

  const float* __restrict__ W = (mat == 0) ? WQ : ((mat == 1) ? WK : WV);
  float* __restrict__ Out = (mat == 0) ? Qo : ((mat == 1) ? Ko : Vo);

  v8f acc[4];
#pragma unroll
  for (int t = 0; t < 4; ++t) acc[t] = vzero8();

  const float* arow = x + (size_t)(m0 + lo) * D_MODEL;

#pragma unroll 8
  for (int k = 0; k < D_MODEL; k += 4) {
    const int kk = k + 2 * hi;
    v2f a = *(const v2f*)(arow + kk);
#pragma unroll
    for (int t = 0; t < 4; ++t) {
      v2f b = *(const v2f*)(W + (size_t)(n0 + t * 16 + lo) * D_MODEL + kk);
      acc[t] = wmma_f32(a, b, acc[t]);
    }
  }

  const int bidx = m0 >> 11;            // SEQ = 2048 rows per batch
  if (mat == 2) {
    // V: plain store into [B, H, S, DK]
#pragma unroll
    for (int t = 0; t < 4; ++t) {
      const int ng = n0 + t * 16 + lo;
      const int h = ng >> 7, d = ng & 127;
#pragma unroll
      for (int i = 0; i < 8; ++i) {
        const int s = (m0 + i + 8 * hi) & (SEQ - 1);
        Out[(((size_t)bidx * NHEADS + h) * SEQ + s) * DK + d] = acc[t][i];
      }
    }
  } else {
    // Q/K: fused RoPE. Pair (2j, 2j+1) sits in adjacent lanes of the C layout.
#pragma unroll
    for (int t = 0; t < 4; ++t) {
      const int ng = n0 + t * 16 + lo;
      const int h = ng >> 7, d = ng & 127;
      const float inv_freq = expf(-(float)(d & ~1) * (LOG_THETA / (float)DK));
      const int odd = d & 1;
#pragma unroll
      for (int i = 0; i < 8; ++i) {
        const int s = (m0 + i + 8 * hi) & (SEQ - 1);
        const float p = (float)tpos[bidx * SEQ + s];
        float sn, cs;
        sincosf(p * inv_freq, &sn, &cs);
        const float val = acc[t][i];
        const float part = __shfl_xor(val, 1, 32);
        // even lane: x1*cos - x2*sin ; odd lane: x1*sin + x2*cos
        const float r = odd ? fmaf(part, sn, val * cs)
                            : fmaf(val, cs, -part * sn);
        Out[(((size_t)bidx * NHEADS + h) * SEQ + s) * DK + d] = r;
      }
    }
  }
}

// =====================================================================
// Kernel 2: causal flash attention, one wave per (b, h, 16-row q tile).
//   scores = Q K^T / sqrt(DK); online softmax; O += P V.
//   P is re-distributed from C layout to A layout through LDS (same-wave
//   DS ops are in-order on CDNA5, no barrier needed in a 1-wave block).
//   grid = (SEQ/16, NHEADS, BATCH), block = 32.
// =====================================================================
__global__ __launch_bounds__(32)
void mha_attn(const float* __restrict__ Q, const float* __restrict__ K,
              const float* __restrict__ V, float* __restrict__ AO) {
  const int lane = threadIdx.x;
  const int hi = lane >> 4;
  const int lo = lane & 15;
  const int qt = blockIdx.x;
  const int h = blockIdx.y;
  const int b = blockIdx.z;

  const size_t head = ((size_t)b * NHEADS + h) * SEQ;
  const float* __restrict__ Qb = Q + head * DK;
  const float* __restrict__ Kb = K + head * DK;
  const float* __restrict__ Vb = V + head * DK;

  __shared__ float Pst[16 * 16];

  // Preload all Q fragments for this 16-row tile: 32 k-steps of 4.
  v2f qa[32];
  const float* qrow = Qb + (size_t)(qt * 16 + lo) * DK;
#pragma unroll
  for (int ks = 0; ks < 32; ++ks)
    qa[ks] = *(const v2f*)(qrow + 4 * ks + 2 * hi);

  float mrow[8], lrow[8], alph[8];
  v8f o[8];
#pragma unroll
  for (int i = 0; i < 8; ++i) {
    mrow[i] = -__builtin_inff();
    lrow[i] = 0.f;
    o[i] = vzero8();
  }

  const float scale = 0.08838834764831845f;  // 1/sqrt(128)

  for (int kt = 0; kt <= qt; ++kt) {
    // ---- scores tile: 32 WMMAs over DK=128 ----
    v8f sc = vzero8();
    const float* krow = Kb + (size_t)(kt * 16 + lo) * DK;
#pragma unroll
    for (int ks = 0; ks < 32; ++ks) {
      v2f kb = *(const v2f*)(krow + 4 * ks + 2 * hi);
      sc = wmma_f32(qa[ks], kb, sc);
    }

    const bool diag = (kt == qt);
    // ---- online softmax (row reductions across the 16-lane half) ----
#pragma unroll
    for (int i = 0; i < 8; ++i) {
      float s = sc[i] * scale;
      if (diag && lo > i + 8 * hi) s = -__builtin_inff();  // causal mask
      float rm = s;
      rm = fmaxf(rm, __shfl_xor(rm, 1, 32));
      rm = fmaxf(rm, __shfl_xor(rm, 2, 32));
      rm = fmaxf(rm, __shfl_xor(rm, 4, 32));
      rm = fmaxf(rm, __shfl_xor(rm, 8, 32));
      const float mnew = fmaxf(mrow[i], rm);
      const float a = expf(mrow[i] - mnew);   // first tile: exp(-inf)=0
      const float p = expf(s - mnew);
      float rs = p;
      rs += __shfl_xor(rs, 1, 32);
      rs += __shfl_xor(rs, 2, 32);
      rs += __shfl_xor(rs, 4, 32);
      rs += __shfl_xor(rs, 8, 32);
      lrow[i] = lrow[i] * a + rs;
      mrow[i] = mnew;
      alph[i] = a;
      Pst[(i + 8 * hi) * 16 + lo] = p;        // stage P in C layout
    }
    // rescale O accumulator rows
#pragma unroll
    for (int t = 0; t < 8; ++t)
#pragma unroll
      for (int i = 0; i < 8; ++i) o[t][i] *= alph[i];

    // ---- O += P(16x16) @ V(16x128): read P back in A layout ----
#pragma unroll
    for (int ks = 0; ks < 4; ++ks) {
      v2f a = *(const v2f*)(&Pst[lo * 16 + 4 * ks + 2 * hi]);
      const float* vrow = Vb + (size_t)(kt * 16 + 4 * ks + 2 * hi) * DK;
#pragma unroll
      for (int t = 0; t < 8; ++t) {
        v2f bv;
        bv[0] = vrow[t * 16 + lo];
        bv[1] = vrow[DK + t * 16 + lo];
        o[t] = wmma_f32(a, bv, o[t]);
      }
    }
  }

  // ---- epilogue: normalize and scatter to [B, S, D_MODEL] ----
#pragma unroll
  for (int i = 0; i < 8; ++i) {
    const float invl = 1.f / lrow[i];
    const size_t row = (size_t)b * SEQ + qt * 16 + i + 8 * hi;
#pragma unroll
    for (int t = 0; t < 8; ++t)
      AO[row * D_MODEL + h * DK + t * 16 + lo] = o[t][i] * invl;
  }
}

// =====================================================================
// Kernel 3: output projection  out = AO @ WO^T  (same NT GEMM shape).
//   grid = (MROWS/16, (D_MODEL/16)/4), block = 32.
// =====================================================================
__global__ __launch_bounds__(32)
void mha_out_proj(const float* __restrict__ A, const float* __restrict__ WO,
                  float* __restrict__ out) {
  const int lane = threadIdx.x;
  const int hi = lane >> 4;
  const int lo = lane & 15;
  const int m0 = blockIdx.x * 16;
  const int n0 = blockIdx.y * 64;

  v8f acc[4];
#pragma unroll
  for (int t = 0; t < 4; ++t) acc[t] = vzero8();

  const float* arow = A + (size_t)(m0 + lo) * D_MODEL;
#pragma unroll 8
  for (int k = 0; k < D_MODEL; k += 4) {
    const int kk = k + 2 * hi;
    v2f a = *(const v2f*)(arow + kk);
#pragma unroll
    for (int t = 0; t < 4; ++t) {
      v2f b = *(const v2f*)(WO + (size_t)(n0 + t * 16 + lo) * D_MODEL + kk);
      acc[t] = wmma_f32(a, b, acc[t]);
    }
  }

#pragma unroll
  for (int t = 0; t < 4; ++t)
#pragma unroll
    for (int i = 0; i < 8; ++i)
      out[(size_t)(m0 + i + 8 * hi) * D_MODEL + n0 + t * 16 + lo] = acc[t][i];
}

// =====================================================================
extern "C" void kernel_launch(void* const* d_in, const int* in_sizes, int n_in,
                              void* d_out, int out_size, void* d_ws, size_t ws_size,
                              hipStream_t stream) {
  (void)in_sizes; (void)n_in; (void)out_size; (void)ws_size;
  const float* x   = (const float*)d_in[0];
  const int*   tp  = (const int*)d_in[1];
  const float* WQ  = (const float*)d_in[2];
  const float* WK  = (const float*)d_in[3];
  const float* WV  = (const float*)d_in[4];
  const float* WO  = (const float*)d_in[5];
  float* out = (float*)d_out;

  const size_t QSZ = (size_t)BATCH * NHEADS * SEQ * DK;  // 8.4M floats
  float* ws = (float*)d_ws;
  float* Qw = ws;
  float* Kw = ws + QSZ;
  float* Vw = ws + 2 * QSZ;
  float* AO = ws + 3 * QSZ;   // total 128 MiB of workspace

  dim3 gProj(MROWS / 16, (D_MODEL / 16) / 4, 3);
  mha_qkv_rope<<<gProj, 32, 0, stream>>>(x, tp, WQ, WK, WV, Qw, Kw, Vw);

  dim3 gAttn(SEQ / 16, NHEADS, BATCH);
  mha_attn<<<gAttn, 32, 0, stream>>>(Qw, Kw, Vw, AO);

  dim3 gOut(MROWS / 16, (D_MODEL / 16) / 4);
  mha_out_proj<<<gOut, 32, 0, stream>>>(AO, WO, out);
}